// TrustMemory_22917945492166
// MI455X (gfx1250) — compile-verified
//
#include <hip/hip_runtime.h>
#include <math.h>

// ---------------------------------------------------------------------------
// TrustMemory leaky-integrator scan, MI455X (gfx1250, wave32).
//   grid = B (32 blocks, one batch row each), block = 256 threads (8 waves)
//   mem state: 4 channels/thread in registers
//   data path: TDM tensor_load_to_lds double-buffered 16xD fp32 tiles,
//              TDM pad feature inserts 1 dword per 256 -> row stride 1028
//              dwords (bank rotation for WMMA row-major A fetches)
//   gate logits: V_WMMA_F32_16X16X4_F32 GEMV per 16-step chunk
// ---------------------------------------------------------------------------

#define B_        32
#define T_        2048
#define D_        1024
#define CHUNK     16
#define NTHREADS  256
#define NWAVES    8
#define KPW       (D_ / NWAVES)      // 128 K-values per wave for WMMA GEMV
#define ROWSTRIDE 1028               // dwords: 1024 + 4 pad dwords (1 per 256)

#if __has_builtin(__builtin_amdgcn_tensor_load_to_lds) && __has_builtin(__builtin_amdgcn_s_wait_tensorcnt)
#define USE_TDM 1
#else
#define USE_TDM 0
#endif
#if __has_builtin(__builtin_amdgcn_wmma_f32_16x16x4_f32)
#define USE_WMMA 1
#else
#define USE_WMMA 0
#endif

typedef __attribute__((ext_vector_type(2))) float        v2f;
typedef __attribute__((ext_vector_type(8))) float        v8f;
typedef __attribute__((ext_vector_type(4))) unsigned int u32x4;
typedef __attribute__((ext_vector_type(8))) int          i32x8;
typedef __attribute__((ext_vector_type(4))) int          i32x4;

__device__ __forceinline__ int ldsCol(int col) { return col + (col >> 8); }

__device__ __forceinline__ float softplusf(float x) {
  return (x > 0.f) ? (x + log1pf(__expf(-x))) : log1pf(__expf(x));
}

__global__ void __launch_bounds__(NTHREADS)
trust_memory_kernel(const float* __restrict__ ws,
                    const unsigned char* __restrict__ pad,
                    const float* __restrict__ projw,
                    const float* __restrict__ projb_p,
                    const float* __restrict__ basem_p,
                    const float* __restrict__ dynm_p,
                    float* __restrict__ out)
{
  extern __shared__ float smem[];
  float* buf0   = smem;                         // CHUNK*ROWSTRIDE
  float* buf1   = buf0 + CHUNK * ROWSTRIDE;     // CHUNK*ROWSTRIDE
  float* wl     = buf1 + CHUNK * ROWSTRIDE;     // D_ (proj_w staged)
  float* plog   = wl + D_;                      // NWAVES*16 partial logits
  float* logits = plog + NWAVES * 16;           // 16
  float* red    = logits + 16;                  // NWAVES*3 reduction slots

  const int b    = blockIdx.x;
  const int tid  = threadIdx.x;
  const int lane = tid & 31;
  const int wave = tid >> 5;

  float* out_bias  = out;
  float* out_mem   = out + (size_t)B_ * T_;
  float* out_trace = out_mem + (size_t)B_ * D_;
  float* out_stab  = out_trace + (size_t)B_ * T_ * D_;

  const float projb = projb_p[0];
  const float base  = softplusf(basem_p[0]);
  const float dyn   = softplusf(dynm_p[0]);

  // bias_stack = ones; stage proj_w into LDS
  for (int i = tid; i < T_; i += NTHREADS) out_bias[(size_t)b * T_ + i] = 1.0f;
  for (int i = tid; i < D_; i += NTHREADS) wl[i] = projw[i];

  const float* wsB = ws + (size_t)b * T_ * D_;

#if !USE_WMMA
  const float w0 = projw[tid * 4 + 0];
  const float w1 = projw[tid * 4 + 1];
  const float w2 = projw[tid * 4 + 2];
  const float w3 = projw[tid * 4 + 3];
#endif

  // --- async tile fetch: TDM descriptor load (or plain-copy fallback) ------
  auto issue_tile = [&](int c, float* dst) {
#if USE_TDM
    if (wave == 0) {
      unsigned long long ga =
          (unsigned long long)(const void*)(wsB + (size_t)c * CHUNK * D_);
      unsigned la = (unsigned)(unsigned long long)
          (__attribute__((address_space(3))) void*)dst;
      // D# group0: count=1 | lds_addr | global_addr | type=2 ("image")
      u32x4 g0 = { 1u,
                   la,
                   (unsigned)ga,
                   (unsigned)((ga >> 32) & 0x01FFFFFFu) | (2u << 30) };
      // D# group1: data_size=4B, pad_enable, pad_interval=256dw, pad_amount=1dw
      //   tensor_dim0=1024, tensor_dim1=16, tile_dim0=1024, tile_dim1=16,
      //   tensor_dim0_stride=1024
      i32x8 g1 = { (int)((2u << 16) | (1u << 20) | (7u << 22)),
                   (int)((unsigned)D_ << 16),          // tensor_dim0 lo16
                   (int)((unsigned)CHUNK << 16),       // tensor_dim1 lo16
                   (int)((unsigned)D_ << 16),          // tile_dim0
                   (int)CHUNK,                         // tile_dim1 (tile_dim2=0)
                   (int)D_,                            // dim0_stride lo32
                   0, 0 };
      i32x4 gz4 = { 0, 0, 0, 0 };
      i32x8 gz8 = { 0, 0, 0, 0, 0, 0, 0, 0 };
      // clang-23 / therock signature: 6 args (g0, g1, g2, g3, extra, cpol)
      __builtin_amdgcn_tensor_load_to_lds(g0, g1, gz4, gz4, gz8, 0);
    }
#else
    for (int r = 0; r < CHUNK; ++r) {
      const float* g = wsB + ((size_t)c * CHUNK + r) * D_ + tid * 4;
      float* l = dst + r * ROWSTRIDE + ldsCol(tid * 4);
      l[0] = g[0]; l[1] = g[1]; l[2] = g[2]; l[3] = g[3];
    }
#endif
  };

  auto wait_tile = [&]() {
#if USE_TDM
    if (wave == 0) __builtin_amdgcn_s_wait_tensorcnt(0);
#endif
    __syncthreads();
  };

  issue_tile(0, buf0);
  __syncthreads();   // proj_w staged in LDS

  float m0 = 0.f, m1 = 0.f, m2 = 0.f, m3 = 0.f;  // resident memory state
  float run_count = 0.f, run_sum = 0.f, run_sq = 0.f;  // redundant per-thread

  for (int c = 0; c < T_ / CHUNK; ++c) {
    float* cur = (c & 1) ? buf1 : buf0;
    wait_tile();
    if (c + 1 < T_ / CHUNK) issue_tile(c + 1, (c & 1) ? buf0 : buf1);

#if USE_WMMA
    // ---- gate logits for the 16 timesteps: fp32 WMMA GEMV, K split by wave
    {
      const int row  = lane & 15;
      const int koff = (lane >> 4) << 1;        // lanes 0-15: K+0/1, 16-31: K+2/3
      const float* arow = cur + row * ROWSTRIDE;
      v8f acc = { 0.f, 0.f, 0.f, 0.f, 0.f, 0.f, 0.f, 0.f };
      const int kb = wave * KPW;
#pragma unroll 4
      for (int k0 = 0; k0 < KPW; k0 += 4) {
        const int k  = kb + k0 + koff;
        const int kc = ldsCol(k);
        v2f a;  a.x  = arow[kc];  a.y  = arow[kc + 1];
        v2f bb; bb.x = wl[k];     bb.y = wl[k + 1];     // B cols replicated
        acc = __builtin_amdgcn_wmma_f32_16x16x4_f32(
            false, a, false, bb, (short)0, acc, false, false);
      }
      union { v8f v; float f[8]; } u; u.v = acc;
      if (lane == 0) {
#pragma unroll
        for (int r = 0; r < 8; ++r) plog[wave * 16 + r] = u.f[r];
      } else if (lane == 16) {
#pragma unroll
        for (int r = 0; r < 8; ++r) plog[wave * 16 + 8 + r] = u.f[r];
      }
    }
    __syncthreads();
    if (tid < 16) {
      float s = projb;
#pragma unroll
      for (int w = 0; w < NWAVES; ++w) s += plog[w * 16 + tid];
      logits[tid] = s;
    }
    __syncthreads();
#endif

    // ---- 16 sequential scan steps over this tile -------------------------
    for (int t = 0; t < CHUNK; ++t) {
      const int tg = c * CHUNK + t;
      const float* rowp = cur + t * ROWSTRIDE + ldsCol(tid * 4);
      const float x0 = rowp[0], x1 = rowp[1], x2 = rowp[2], x3 = rowp[3];

      const float e0 = x0 - m0, e1 = x1 - m1, e2 = x2 - m2, e3 = x3 - m3;
      float p1 = e0 * e0 + e1 * e1 + e2 * e2 + e3 * e3;   // ||ws - mem||^2 part
      float p2 = m0 * m0 + m1 * m1 + m2 * m2 + m3 * m3;   // ||mem||^2 part
#if !USE_WMMA
      float p3 = x0 * w0 + x1 * w1 + x2 * w2 + x3 * w3;   // dot(ws, proj_w)
#endif
#pragma unroll
      for (int off = 16; off > 0; off >>= 1) {
        p1 += __shfl_xor(p1, off, 32);
        p2 += __shfl_xor(p2, off, 32);
#if !USE_WMMA
        p3 += __shfl_xor(p3, off, 32);
#endif
      }
      if (lane == 0) {
        red[wave * 3 + 0] = p1;
        red[wave * 3 + 1] = p2;
#if !USE_WMMA
        red[wave * 3 + 2] = p3;
#endif
      }
      __syncthreads();

      float s1 = 0.f, s2 = 0.f;
#if !USE_WMMA
      float s3 = 0.f;
#endif
#pragma unroll
      for (int w = 0; w < NWAVES; ++w) {
        s1 += red[w * 3 + 0];
        s2 += red[w * 3 + 1];
#if !USE_WMMA
        s3 += red[w * 3 + 2];
#endif
      }
#if USE_WMMA
      const float logit = logits[t];
#else
      const float logit = s3 + projb;
#endif
      // gate (all threads redundantly; pad_mask load is a broadcast)
      const float adaptive = 1.f / (1.f + __expf(-logit));
      const float eff = fminf(fmaxf(0.01f * (base + dyn * adaptive), 1e-6f), 0.95f);
      const float valid = pad[(size_t)b * T_ + tg] ? 0.f : 1.f;
      const float g = eff * valid;

      // stability statistics (redundant scalar chain in every thread)
      const float delta = sqrtf(g * g * s1 + 1e-12f);
      const float bn    = fmaxf(sqrtf(s2 + 1e-12f), 1e-6f);
      const float ratio = delta / bn;
      const float reff  = (run_count == 0.f && valid > 0.f) ? 0.f : ratio;
      run_count += valid;
      run_sum   += reff * valid;
      run_sq    += reff * reff * valid;
      const float sc = fmaxf(run_count, 1.f);
      const float cm = run_sum / sc;
      const float cv = fmaxf(run_sq / sc - cm * cm, 0.f);
      const float stab = __expf(-(cm + sqrtf(cv + 1e-8f)));
      if (tid == 0)
        out_stab[(size_t)b * T_ + tg] = (valid > 0.f) ? stab : 1.0f;
      __syncthreads();   // red[] reuse guard before next step

      // state update + trace store (coalesced b128)
      const float ia = 1.f - g;
      m0 = ia * m0 + g * x0;
      m1 = ia * m1 + g * x1;
      m2 = ia * m2 + g * x2;
      m3 = ia * m3 + g * x3;
      *(float4*)(out_trace + ((size_t)b * T_ + tg) * D_ + tid * 4) =
          make_float4(m0, m1, m2, m3);
    }
  }

  *(float4*)(out_mem + (size_t)b * D_ + tid * 4) = make_float4(m0, m1, m2, m3);
}

extern "C" void kernel_launch(void* const* d_in, const int* in_sizes, int n_in,
                              void* d_out, int out_size, void* d_ws, size_t ws_size,
                              hipStream_t stream) {
  (void)in_sizes; (void)n_in; (void)d_ws; (void)ws_size; (void)out_size;
  const float*         ws_in = (const float*)d_in[0];
  const unsigned char* padm  = (const unsigned char*)d_in[1];  // jax bool -> u8
  const float*         projw = (const float*)d_in[2];
  const float*         projb = (const float*)d_in[3];
  const float*         basem = (const float*)d_in[4];
  const float*         dynm  = (const float*)d_in[5];

  const size_t shmem =
      (size_t)(2 * CHUNK * ROWSTRIDE + D_ + NWAVES * 16 + 16 + NWAVES * 3) *
      sizeof(float);

  trust_memory_kernel<<<dim3(B_), dim3(NTHREADS), shmem, stream>>>(
      ws_in, padm, projw, projb, basem, dynm, (float*)d_out);
}